// ObjectDetector_59287728554608
// MI455X (gfx1250) — compile-verified
//
#include <hip/hip_runtime.h>
#include <hip/hip_bf16.h>
#include <stdint.h>

// ---------------- constants from the reference ----------------
#define IMG_W 1024.0f
#define IMG_H 1024.0f
#define CONF_THRESH 0.2f
#define NMS_THRESH 0.5f
#define MAX_PRED 300
#define MAX_DELTA 5.0f

#define KPAD 304            // 300 padded to 19*16 for WMMA tiles
#define NTILE 19            // 304/16
#define CAND_CAP 1024       // gather capacity before bitonic sort
#define TILE_KEYS 4096      // keys per async LDS tile (16KB)

typedef __attribute__((ext_vector_type(2))) float v2f;
typedef __attribute__((ext_vector_type(8))) float v8f;

// ---------------- shared decode helper ----------------
__device__ __forceinline__ void decode_box(const float4 a, const float4 d,
                                           float& x1, float& y1, float& x2, float& y2,
                                           bool& valid, bool& size_ok) {
    float w  = a.z - a.x;
    float h  = a.w - a.y;
    float cx = a.x + 0.5f * w;
    float cy = a.y + 0.5f * h;
    float d0 = fminf(fmaxf(d.x, -MAX_DELTA), MAX_DELTA);
    float d1 = fminf(fmaxf(d.y, -MAX_DELTA), MAX_DELTA);
    float d2 = fminf(fmaxf(d.z, -MAX_DELTA), MAX_DELTA);
    float d3 = fminf(fmaxf(d.w, -MAX_DELTA), MAX_DELTA);
    float pcx = cx + d0 * w;
    float pcy = cy + d1 * h;
    float pw  = w * __expf(d2);
    float ph  = h * __expf(d3);
    x1 = fminf(fmaxf(pcx - 0.5f * pw, 0.0f), IMG_W);
    y1 = fminf(fmaxf(pcy - 0.5f * ph, 0.0f), IMG_H);
    x2 = fminf(fmaxf(pcx + 0.5f * pw, 0.0f), IMG_W);
    y2 = fminf(fmaxf(pcy + 0.5f * ph, 0.0f), IMG_H);
    float bw = x2 - x1;
    float bh = y2 - y1;
    valid   = (bw > 1.0f) & (bh > 1.0f) & (bw < 2000.0f) & (bh < 2000.0f);
    size_ok = (bw > 5.0f) & (bh > 5.0f) & (bw < IMG_W * 0.5f) & (bh < IMG_H * 0.5f);
}

__device__ __forceinline__ unsigned score_key(float logit, const float4 a, const float4 d) {
    float x1, y1, x2, y2; bool valid, size_ok;
    decode_box(a, d, x1, y1, x2, y2, valid, size_ok);
    float s = 1.0f / (1.0f + __expf(-logit));
    bool conf = (s > CONF_THRESH) & valid & size_ok;
    return conf ? __float_as_uint(s) : 0u;
}

// ======================================================================
// Kernel 1: streaming decode + score -> sortable key per (b, anchor).
// Fully vectorized: each thread does 4 anchors (b128 loads/stores).
// Bandwidth bound: ~110 MB in, 21 MB out  ->  ~6 us at 23.3 TB/s.
// ======================================================================
__global__ void __launch_bounds__(256)
k_decode_score(const float4* __restrict__ logits4,
               const float4* __restrict__ deltas4,
               const float4* __restrict__ anchors4,
               uint4* __restrict__ keys4,
               int N) {
    const int b  = blockIdx.y;
    const int n4 = N >> 2;
    const int g  = blockIdx.x * blockDim.x + threadIdx.x;   // group of 4 anchors
    if (g >= n4) return;
    float4 lg = logits4[(size_t)b * n4 + g];
    const int i0 = g << 2;
    unsigned k0 = score_key(lg.x, anchors4[i0 + 0], deltas4[(size_t)b * N + i0 + 0]);
    unsigned k1 = score_key(lg.y, anchors4[i0 + 1], deltas4[(size_t)b * N + i0 + 1]);
    unsigned k2 = score_key(lg.z, anchors4[i0 + 2], deltas4[(size_t)b * N + i0 + 2]);
    unsigned k3 = score_key(lg.w, anchors4[i0 + 3], deltas4[(size_t)b * N + i0 + 3]);
    keys4[(size_t)b * n4 + g] = make_uint4(k0, k1, k2, k3);
}

// ======================================================================
// Async LDS staging for the key scans (GLOBAL_LOAD_ASYNC_TO_LDS_B128,
// ASYNCcnt / s_wait_asynccnt): each of 1024 threads copies 16B per tile.
// ======================================================================
__device__ __forceinline__ void issue_async_tile(const unsigned* __restrict__ gbase,
                                                 unsigned elemOff,
                                                 unsigned* ldsbuf, int t) {
    unsigned ldsaddr = (unsigned)(uintptr_t)(ldsbuf + t * 4);       // low 32b = LDS byte addr
    unsigned voff    = (elemOff + (unsigned)t * 4u) * 4u;           // byte offset from base
    asm volatile("global_load_async_to_lds_b128 %0, %1, %2"
                 :: "v"(ldsaddr), "v"(voff), "s"(gbase) : "memory");
}

template <typename F>
__device__ __forceinline__ void scan_keys_async(const unsigned* __restrict__ kb, int N,
                                                unsigned* stage /* [2*TILE_KEYS] */,
                                                F f) {
    const int t  = threadIdx.x;            // 0..1023
    const int nt = N / TILE_KEYS;          // full tiles
    if (nt > 0) {
        issue_async_tile(kb, 0u, stage, t);
        for (int tile = 0; tile < nt; ++tile) {
            if (tile + 1 < nt) {
                issue_async_tile(kb, (unsigned)(tile + 1) * TILE_KEYS,
                                 stage + ((tile + 1) & 1) * TILE_KEYS, t);
                asm volatile("s_wait_asynccnt 0x1" ::: "memory");   // tile `tile` landed
            } else {
                asm volatile("s_wait_asynccnt 0x0" ::: "memory");
            }
            __syncthreads();
            const unsigned* tb = stage + (tile & 1) * TILE_KEYS + t * 4;
            const int base = tile * TILE_KEYS + t * 4;
            #pragma unroll
            for (int k = 0; k < 4; ++k) f(tb[k], base + k);
            __syncthreads();               // all readers done before buffer reuse
        }
    }
    for (int i = nt * TILE_KEYS + t; i < N; i += 1024) f(kb[i], i);  // tail
}

// ======================================================================
// Kernel 2: per-batch top-300 via 2-level radix select + bitonic sort.
// One 1024-thread workgroup per batch.
// ======================================================================
__global__ void __launch_bounds__(1024)
k_select_topk(const unsigned* __restrict__ keys,
              unsigned* __restrict__ sel_idx,
              float* __restrict__ sel_score,
              int N) {
    __shared__ unsigned hist[4096];
    __shared__ __align__(16) unsigned stage[2 * TILE_KEYS];
    __shared__ unsigned ckey[CAND_CAP];
    __shared__ unsigned cidx[CAND_CAP];
    __shared__ unsigned sh_t1, sh_above, sh_thresh, sh_cnt, sh_done;

    const int t = threadIdx.x;
    const int b = blockIdx.x;
    const unsigned* kb = keys + (size_t)b * N;

    // ---- level 1: histogram of top 12 bits ----
    for (int i = t; i < 4096; i += 1024) hist[i] = 0;
    if (t == 0) { sh_cnt = 0; sh_done = 0; }
    __syncthreads();
    scan_keys_async(kb, N, stage, [&](unsigned k, int) {
        if (k) atomicAdd(&hist[k >> 20], 1u);
    });
    __syncthreads();
    if (t == 0) {
        unsigned cum = 0; int t1 = -1; unsigned above = 0;
        for (int bin = 4095; bin >= 0; --bin) {
            unsigned c = hist[bin];
            if (cum + c >= MAX_PRED) { t1 = bin; above = cum; break; }
            cum += c;
        }
        if (t1 < 0) { sh_thresh = 1u; sh_done = 1u; }   // < 300 candidates: take all nonzero
        else        { sh_t1 = (unsigned)t1; sh_above = above; }
    }
    __syncthreads();

    // ---- level 2: refine within boundary bucket (bits 19:8) ----
    if (!sh_done) {
        const unsigned t1 = sh_t1;
        for (int i = t; i < 4096; i += 1024) hist[i] = 0;
        __syncthreads();
        scan_keys_async(kb, N, stage, [&](unsigned k, int) {
            if (k && (k >> 20) == t1) atomicAdd(&hist[(k >> 8) & 0xFFFu], 1u);
        });
        __syncthreads();
        if (t == 0) {
            unsigned need = MAX_PRED - sh_above;
            unsigned cum = 0; unsigned t2 = 0;
            for (int bin = 4095; bin >= 0; --bin) {
                unsigned c = hist[bin];
                if (cum + c >= need) { t2 = (unsigned)bin; break; }
                cum += c;
            }
            unsigned thr = ((t1 << 12) | t2) << 8;
            sh_thresh = thr ? thr : 1u;
        }
        __syncthreads();
    }

    // ---- gather candidates >= threshold ----
    const unsigned thr = sh_thresh;
    scan_keys_async(kb, N, stage, [&](unsigned k, int i) {
        if (k && k >= thr) {
            unsigned p = atomicAdd(&sh_cnt, 1u);
            if (p < CAND_CAP) { ckey[p] = k; cidx[p] = (unsigned)i; }
        }
    });
    __syncthreads();
    const unsigned cnt = sh_cnt < CAND_CAP ? sh_cnt : CAND_CAP;
    if ((unsigned)t >= cnt) { ckey[t] = 0u; cidx[t] = 0x7FFFFFFFu; }

    // ---- bitonic sort 1024 pairs: key desc, idx asc on ties ----
    for (unsigned k = 2; k <= CAND_CAP; k <<= 1) {
        for (unsigned j = k >> 1; j > 0; j >>= 1) {
            __syncthreads();
            unsigned ixj = (unsigned)t ^ j;
            if (ixj > (unsigned)t) {
                unsigned ka = ckey[t],  kc = ckey[ixj];
                unsigned ia = cidx[t],  ic = cidx[ixj];
                bool aFirst = (ka > kc) || (ka == kc && ia < ic);
                bool desc = (((unsigned)t & k) == 0);
                if (desc != aFirst) {
                    ckey[t] = kc; ckey[ixj] = ka;
                    cidx[t] = ic; cidx[ixj] = ia;
                }
            }
        }
    }
    __syncthreads();

    if (t < MAX_PRED) {
        unsigned k = ckey[t];
        sel_idx[b * MAX_PRED + t]   = cidx[t];
        sel_score[b * MAX_PRED + t] = k ? __uint_as_float(k) : -1.0f;
    }
}

// ======================================================================
// Kernel 3: per-batch NMS + output. One workgroup of 320 threads (10 waves).
// area_i + area_j computed per 16x16 tile via v_wmma_f32_16x16x4_f32:
//   A row i = (area_i, 1, 0, 0), B col j = (1, area_j, 0, 0)  =>  C = area_i + area_j
// ======================================================================
__global__ void __launch_bounds__(320)
k_nms_out(const float4* __restrict__ deltas4,
          const float4* __restrict__ anchors4,
          const unsigned* __restrict__ sel_idx,
          const float* __restrict__ sel_score,
          float* __restrict__ out,
          int N) {
    __shared__ float sx1[KPAD], sy1[KPAD], sx2[KPAD], sy2[KPAD], sarea[KPAD], ssc[KPAD];
    __shared__ unsigned smask[MAX_PRED * 10];   // per-row suppression bitmask (j < i only)
    __shared__ unsigned skeep[10];

    const int t = threadIdx.x;
    const int b = blockIdx.x;

    // load + re-decode the 300 survivors
    if (t < KPAD) {
        float x1 = 0, y1 = 0, x2 = 0, y2 = 0, sc = -1.0f;
        if (t < MAX_PRED) {
            sc = sel_score[b * MAX_PRED + t];
            unsigned idx = sel_idx[b * MAX_PRED + t];
            if (sc > 0.0f && idx < (unsigned)N) {
                float4 a = anchors4[idx];
                float4 d = deltas4[(size_t)b * N + idx];
                bool v, so;
                decode_box(a, d, x1, y1, x2, y2, v, so);
            }
        }
        sx1[t] = x1; sy1[t] = y1; sx2[t] = x2; sy2[t] = y2;
        sarea[t] = (x2 - x1) * (y2 - y1);
        ssc[t] = sc;
    }
    for (int w = t; w < MAX_PRED * 10; w += 320) smask[w] = 0u;
    __syncthreads();

    // tiled pairwise pass: 19x19 tiles of 16x16, distributed across 10 waves
    const int wave = t >> 5;
    const int lane = t & 31;
    for (int tid = wave; tid < NTILE * NTILE; tid += 10) {
        int ti = tid / NTILE, tj = tid % NTILE;
        int m = lane & 15;
        // A fragment (16x4 f32): lanes 0-15 hold K=0,1 ; lanes 16-31 hold K=2,3 (zeros)
        v2f af, bf;
        af[0] = (lane < 16) ? sarea[ti * 16 + m] : 0.0f;
        af[1] = (lane < 16) ? 1.0f : 0.0f;
        // B fragment (4x16 f32): lanes 0-15 hold rows K=0,1 ; lanes 16-31 rows K=2,3 (zeros)
        bf[0] = (lane < 16) ? 1.0f : 0.0f;
        bf[1] = (lane < 16) ? sarea[tj * 16 + m] : 0.0f;
        v8f c = {};
        c = __builtin_amdgcn_wmma_f32_16x16x4_f32(false, af, false, bf,
                                                  (short)0, c, false, false);
        // C layout: VGPR r, lane l -> M = r + (l>=16 ? 8 : 0), Nc = l & 15
        int jj = tj * 16 + (lane & 15);
        #pragma unroll
        for (int r = 0; r < 8; ++r) {
            int ii = ti * 16 + r + ((lane >= 16) ? 8 : 0);
            if (ii < MAX_PRED && jj < ii) {
                float ix1 = fmaxf(sx1[ii], sx1[jj]);
                float iy1 = fmaxf(sy1[ii], sy1[jj]);
                float ix2 = fminf(sx2[ii], sx2[jj]);
                float iy2 = fminf(sy2[ii], sy2[jj]);
                float inter = fmaxf(ix2 - ix1, 0.0f) * fmaxf(iy2 - iy1, 0.0f);
                float iou = __fdividef(inter, c[r] - inter + 1e-9f);  // fast rcp: threshold test
                if (iou >= NMS_THRESH)
                    atomicOr(&smask[ii * 10 + (jj >> 5)], 1u << (jj & 31));
            }
        }
    }
    __syncthreads();

    // sequential greedy keep (exact reference recurrence), single lane
    if (t == 0) {
        unsigned keepw[10] = {0};
        for (int i = 0; i < MAX_PRED; ++i) {
            unsigned sup = 0;
            #pragma unroll
            for (int w = 0; w < 10; ++w) sup |= smask[i * 10 + w] & keepw[w];
            if (ssc[i] > 0.0f && sup == 0) keepw[i >> 5] |= 1u << (i & 31);
        }
        #pragma unroll
        for (int w = 0; w < 10; ++w) skeep[w] = keepw[w];
    }
    __syncthreads();

    if (t < MAX_PRED) {
        float k = (float)((skeep[t >> 5] >> (t & 31)) & 1u);
        float* o = out + ((size_t)b * MAX_PRED + t) * 5;
        o[0] = sx1[t] * (1.0f / IMG_W) * k;
        o[1] = sy1[t] * (1.0f / IMG_H) * k;
        o[2] = sx2[t] * (1.0f / IMG_W) * k;
        o[3] = sy2[t] * (1.0f / IMG_H) * k;
        o[4] = ssc[t] * k;
    }
}

// ======================================================================
extern "C" void kernel_launch(void* const* d_in, const int* in_sizes, int n_in,
                              void* d_out, int out_size, void* d_ws, size_t ws_size,
                              hipStream_t stream) {
    const float4* logits4 = (const float4*)d_in[0];  // (B, N)
    const float4* deltas4 = (const float4*)d_in[1];  // (B, N, 4)
    const float4* anchors4= (const float4*)d_in[2];  // (N, 4)

    const int N = in_sizes[2] / 4;           // anchors
    const int B = in_sizes[0] / N;           // batches (16)

    uint8_t* ws = (uint8_t*)d_ws;
    size_t keysBytes = (size_t)B * N * sizeof(unsigned);
    unsigned* keys      = (unsigned*)ws;
    unsigned* sel_idx   = (unsigned*)(ws + keysBytes);
    float*    sel_score = (float*)(ws + keysBytes + (size_t)B * MAX_PRED * sizeof(unsigned));
    float*    out       = (float*)d_out;

    dim3 grid1((N / 4 + 255) / 256, B);
    k_decode_score<<<grid1, 256, 0, stream>>>(logits4, deltas4, anchors4, (uint4*)keys, N);
    k_select_topk<<<B, 1024, 0, stream>>>(keys, sel_idx, sel_score, N);
    k_nms_out<<<B, 320, 0, stream>>>(deltas4, anchors4, sel_idx, sel_score, out, N);
}